// CTRNN_69965017251987
// MI455X (gfx1250) — compile-verified
//
#include <hip/hip_runtime.h>

typedef __attribute__((ext_vector_type(2))) float v2f;
typedef __attribute__((ext_vector_type(4))) float v4f;
typedef __attribute__((ext_vector_type(8))) float v8f;
typedef __attribute__((ext_vector_type(4))) int   v4i;

#define CT_ALPHA 0.2f
#define CT_OMA   0.8f
#define KC 32                 // K-chunk depth staged in LDS (phase 1)

// ---- gfx1250 async global->LDS path (guarded; fallback = regs + ds_store) ----
#if defined(__has_builtin)
#  if __has_builtin(__builtin_amdgcn_global_load_async_to_lds_b128)
#    define CT_ASYNC_LDS 1
#  else
#    define CT_ASYNC_LDS 0
#  endif
#else
#  define CT_ASYNC_LDS 0
#endif

#if CT_ASYNC_LDS
// builtin signature (from clang diagnostic): arg0 = v4i addrspace(1)* (global
// src), arg1 = LDS dst, arg2 = imm offset, arg3 = imm cpol
#  define CT_GPTR(p) ((__attribute__((address_space(1))) v4i*)(p))
#  define CT_LPTR(p) ((__attribute__((address_space(3))) v4i*)(p))
#  if __has_builtin(__builtin_amdgcn_s_wait_asynccnt)
#    define CT_WAIT_ASYNC(n) __builtin_amdgcn_s_wait_asynccnt(n)
#  else
#    define CT_WAIT_ASYNC(n) asm volatile("s_wait_asynccnt %0" ::"n"(n) : "memory")
#  endif
#endif

// ---------------------------------------------------------------------------
// Phase 1: out[m,n] = sum_k x[m,k] * W_in[n,k] + b_in[n]
//   Workgroup (256 thr, 8 waves) computes a 128(M) x 64(N) tile.
//   B strip (64 cols x KC k-depth) staged in LDS, double buffered, shared by
//   all 8 waves; copies issued with async-load-to-LDS and overlapped with the
//   WMMAs of the previous chunk.
//
// fp32 WMMA 16x16x4 fragment layout (ISA 7.12.2):
//   A: lanes 0-15 -> M=lane, v0=K0 v1=K1 ; lanes 16-31 -> M=lane-16, K2/K3
//   B: lanes 0-15 -> N=lane, v0=K0 v1=K1 ; lanes 16-31 -> N=lane-16, K2/K3
//   C/D: VGPR r -> row (r or r+8), col = lane & 15
// ---------------------------------------------------------------------------
__global__ __launch_bounds__(256)
void ctrnn_xproj_kernel(const float* __restrict__ x,
                        const float* __restrict__ W_in,
                        const float* __restrict__ b_in,
                        float* __restrict__ out,
                        int M, int N, int K) {
    __shared__ __align__(16) float sB[2][64 * KC];

    const int tid  = threadIdx.x;
    const int lane = tid & 31;
    const int wave = tid >> 5;
    const int nStrips = N >> 6;
    const int mblk = blockIdx.x / nStrips;
    const int ns   = blockIdx.x % nStrips;
    const int m0   = (mblk << 7) + (wave << 4);   // this wave's 16 rows
    const int n0   = ns << 6;                     // 64-col strip
    const int half = lane >> 4;
    const int l    = lane & 15;
    const int koff = half << 1;

    const float* aRow = x + (size_t)(m0 + l) * K + koff;

    // staging map: thread -> cols (c, c+32), 4 consecutive k each (b128)
    const int sc = tid >> 3;                      // 0..31
    const int sk = (tid & 7) << 2;                // 0,4,..,28
    const float* gB0 = W_in + (size_t)(n0 + sc)      * K + sk;
    const float* gB1 = W_in + (size_t)(n0 + sc + 32) * K + sk;
    const int ldsOff0 = sc * KC + sk;
    const int ldsOff1 = (sc + 32) * KC + sk;

    v8f acc0 = {}, acc1 = {}, acc2 = {}, acc3 = {};
    const int nChunks = K / KC;

    // prologue: stage chunk 0 into buffer 0
    {
#if CT_ASYNC_LDS
        __builtin_amdgcn_global_load_async_to_lds_b128(
            CT_GPTR(gB0), CT_LPTR(&sB[0][ldsOff0]), 0, 0);
        __builtin_amdgcn_global_load_async_to_lds_b128(
            CT_GPTR(gB1), CT_LPTR(&sB[0][ldsOff1]), 0, 0);
#else
        *(v4f*)&sB[0][ldsOff0] = *(const v4f*)gB0;
        *(v4f*)&sB[0][ldsOff1] = *(const v4f*)gB1;
#endif
    }

    for (int c = 0; c < nChunks; ++c) {
        const int k0 = c * KC;
        // stage next chunk into the other buffer (overlaps this chunk's math)
        if (c + 1 < nChunks) {
            const int kn = k0 + KC;
            const int nb = (c + 1) & 1;
#if CT_ASYNC_LDS
            __builtin_amdgcn_global_load_async_to_lds_b128(
                CT_GPTR(gB0 + kn), CT_LPTR(&sB[nb][ldsOff0]), 0, 0);
            __builtin_amdgcn_global_load_async_to_lds_b128(
                CT_GPTR(gB1 + kn), CT_LPTR(&sB[nb][ldsOff1]), 0, 0);
#else
            *(v4f*)&sB[nb][ldsOff0] = *(const v4f*)(gB0 + kn);
            *(v4f*)&sB[nb][ldsOff1] = *(const v4f*)(gB1 + kn);
#endif
        }
#if CT_ASYNC_LDS
        if (c + 1 < nChunks) { CT_WAIT_ASYNC(2); }   // chunk c landed, c+1 in flight
        else                 { CT_WAIT_ASYNC(0); }
#endif
        __syncthreads();   // chunk c visible to all waves

        // batch-load this wave's A fragments for the whole chunk (one wait)
        v2f aF[KC / 4];
#pragma unroll
        for (int kk = 0; kk < KC / 4; ++kk)
            aF[kk] = *(const v2f*)(aRow + k0 + (kk << 2));

        const float* bBase = &sB[c & 1][0];
#pragma unroll
        for (int kk = 0; kk < KC / 4; ++kk) {
            const int kb = (kk << 2) + koff;
            v2f b0 = *(const v2f*)(bBase + ( 0 + l) * KC + kb);
            v2f b1 = *(const v2f*)(bBase + (16 + l) * KC + kb);
            v2f b2 = *(const v2f*)(bBase + (32 + l) * KC + kb);
            v2f b3 = *(const v2f*)(bBase + (48 + l) * KC + kb);
            acc0 = __builtin_amdgcn_wmma_f32_16x16x4_f32(false, aF[kk], false, b0,
                                                         (short)0, acc0, false, false);
            acc1 = __builtin_amdgcn_wmma_f32_16x16x4_f32(false, aF[kk], false, b1,
                                                         (short)0, acc1, false, false);
            acc2 = __builtin_amdgcn_wmma_f32_16x16x4_f32(false, aF[kk], false, b2,
                                                         (short)0, acc2, false, false);
            acc3 = __builtin_amdgcn_wmma_f32_16x16x4_f32(false, aF[kk], false, b3,
                                                         (short)0, acc3, false, false);
        }
        __syncthreads();   // all waves done reading buf[c&1] before it is reused
    }

    // epilogue: add bias, scatter 16x64 strip
    const int rbase = m0 + (half << 3);
#pragma unroll
    for (int j = 0; j < 4; ++j) {
        v8f acc = (j == 0) ? acc0 : (j == 1) ? acc1 : (j == 2) ? acc2 : acc3;
        const int col = n0 + (j << 4) + l;
        const float bias = b_in[col];
#pragma unroll
        for (int r = 0; r < 8; ++r)
            out[(size_t)(rbase + r) * N + col] = acc[r] + bias;
    }
}

// ---------------------------------------------------------------------------
// Phase 2 (one launch per timestep, stream-ordered):
//   pre[b,j]   = sum_k h_in[b,k] * W_hh[j,k] + b_hh[j] + xp[b,j]
//   h_new[b,j] = 0.8*h_in[b,j] + 0.2*sigmoid(pre)
// K=1024 split over 4 waves (64-deep WMMA chains instead of 256), partials
// reduced through LDS. Workgroup = 8 waves = 2 tiles x 4 K-chunks.
// K-loop uses depth-1 register rotation so each WMMA only waits on loads
// issued a full iteration earlier.
// ---------------------------------------------------------------------------
__global__ __launch_bounds__(256)
void ctrnn_step_kernel(const float* __restrict__ h_in,
                       const float* __restrict__ W_hh,
                       const float* __restrict__ b_hh,
                       float* __restrict__ xp_out,   // d_out + t*B*H, in-place
                       float* __restrict__ h_out,
                       int H) {
    __shared__ __align__(32) float sRed[8][32][8];

    const int tid  = threadIdx.x;
    const int lane = tid & 31;
    const int wave = tid >> 5;
    const int kc   = wave & 3;                       // K-chunk 0..3
    const int tile = blockIdx.x * 2 + (wave >> 2);   // global 16x16 tile id
    const int nTiles = H >> 4;
    const int mt = tile / nTiles;
    const int nt = tile % nTiles;
    const int m0 = mt << 4;
    const int n0 = nt << 4;
    const int half = lane >> 4;
    const int l    = lane & 15;
    const int koff = half << 1;

    const int kLen = H >> 2;                         // 256
    const int k0   = kc * kLen;
    const int kend = k0 + kLen;

    const float* aRow = h_in + (size_t)(m0 + l) * H + koff;
    const float* bCol = W_hh + (size_t)(n0 + l) * H + koff;

    v8f acc = {};
    v2f a = *(const v2f*)(aRow + k0);
    v2f b = *(const v2f*)(bCol + k0);
#pragma unroll 4
    for (int k = k0; k < kend; k += 4) {
        const int kn = (k + 4 < kend) ? (k + 4) : k;   // scalar-clamped prefetch
        v2f an = *(const v2f*)(aRow + kn);
        v2f bn = *(const v2f*)(bCol + kn);
        acc = __builtin_amdgcn_wmma_f32_16x16x4_f32(false, a, false, b,
                                                    (short)0, acc, false, false);
        a = an; b = bn;
    }

    // cross-wave K reduction through LDS
    *(v8f*)&sRed[wave][lane][0] = acc;
    __syncthreads();
    if (kc != 0) return;

    v8f total = acc;
#pragma unroll
    for (int j = 1; j < 4; ++j)
        total = total + *(const v8f*)&sRed[wave + j][lane][0];

    const int col  = n0 + l;
    const float bias = b_hh[col];
    const int rbase = m0 + (half << 3);
#pragma unroll
    for (int r = 0; r < 8; ++r) {
        const size_t idx = (size_t)(rbase + r) * H + col;
        const float pre = total[r] + bias + xp_out[idx];
        const float s  = 1.0f / (1.0f + __expf(-pre));
        const float hn = CT_OMA * h_in[idx] + CT_ALPHA * s;
        xp_out[idx] = hn;
        h_out[idx]  = hn;
    }
}

// ---------------------------------------------------------------------------
// Inputs (setup_inputs order): x(T,B,I), hidden(B,H), W_in(H,I), b_in(H),
//                              W_hh(H,H), b_hh(H)
// Output: concat(output(T,B,H), h_last(B,H)) fp32.
// ---------------------------------------------------------------------------
extern "C" void kernel_launch(void* const* d_in, const int* in_sizes, int n_in,
                              void* d_out, int out_size, void* d_ws, size_t ws_size,
                              hipStream_t stream) {
    const float* x      = (const float*)d_in[0];
    const float* hidden = (const float*)d_in[1];
    const float* W_in   = (const float*)d_in[2];
    const float* b_in   = (const float*)d_in[3];
    const float* W_hh   = (const float*)d_in[4];
    const float* b_hh   = (const float*)d_in[5];
    float* out = (float*)d_out;

    const int H = in_sizes[3];              // 1024
    const int I = in_sizes[2] / H;          // 512
    const int B = in_sizes[1] / H;          // 64
    const int T = in_sizes[0] / (B * I);    // 512
    const int M = T * B;                    // 32768

    // Phase 1: x_proj -> d_out[0 .. T*B*H). 128x64 tile per WG.
    {
        const int blocks = (M >> 7) * (H >> 6);   // 256 * 16 = 4096
        ctrnn_xproj_kernel<<<blocks, 256, 0, stream>>>(x, W_in, b_in, out, M, H, I);
    }

    // Phase 2: T stream-ordered step kernels, ping-pong h in d_ws.
    float* hbuf0 = (float*)d_ws;
    float* hbuf1 = hbuf0 + (size_t)B * H;
    const int stepBlocks = ((B >> 4) * (H >> 4)) >> 1;   // 256 tiles / 2 per WG = 128
    for (int t = 0; t < T; ++t) {
        const float* hin = (t == 0) ? hidden : ((t & 1) ? hbuf0 : hbuf1);
        float* hout = (t & 1) ? hbuf1 : hbuf0;
        ctrnn_step_kernel<<<stepBlocks, 256, 0, stream>>>(
            hin, W_hh, b_hh, out + (size_t)t * B * H, hout, H);
    }

    // h_last = h after step T-1
    float* hlast = ((T - 1) & 1) ? hbuf1 : hbuf0;
    (void)hipMemcpyAsync(out + (size_t)T * B * H, hlast,
                         (size_t)B * H * sizeof(float),
                         hipMemcpyDeviceToDevice, stream);
}